// LengthRegulator_31207232373112
// MI455X (gfx1250) — compile-verified
//
#include <hip/hip_runtime.h>

typedef float __attribute__((ext_vector_type(2))) v2f;
typedef float __attribute__((ext_vector_type(8))) v8f;

#define B_ 32
#define S_ 512
#define D_ 256
#define F_ 256
#define K_ 3

// ---------------------------------------------------------------------------
// Repack w[F][D][K] -> Wt[K][D/2][F][2]  (pair consecutive d so each WMMA
// B-fragment is a single b64 load per lane: {B[d,f], B[d+1,f]})
// ---------------------------------------------------------------------------
__global__ void repack_w_kernel(const float* __restrict__ W, float* __restrict__ Wt) {
    int idx = blockIdx.x * blockDim.x + threadIdx.x;     // < K*D*F
    if (idx >= K_ * D_ * F_) return;
    int p   = idx & 1;
    int tmp = idx >> 1;
    int f   = tmp % F_;  tmp /= F_;
    int d2  = tmp % (D_ / 2);
    int kk  = tmp / (D_ / 2);
    int d   = 2 * d2 + p;
    Wt[idx] = W[((size_t)f * D_ + d) * K_ + kk];
}

// ---------------------------------------------------------------------------
// Conv1d (K=3, SAME) as WMMA f32 GEMM.
// One wave per 16(s) x 64(f) tile: A fragment reused across 4 f-subtiles.
// A layout (16x4 f32):  lane l (lm=l&15, kh=l>>4): V0=A[lm, 2*kh], V1=A[lm, 2*kh+1]
// B layout (4x16 f32):  lane l: n=lm, V0=B[2*kh, n], V1=B[2*kh+1, n]
// C layout (16x16 f32): VGPR i, lane l: row = i + 8*kh, col = lm
// ---------------------------------------------------------------------------
__global__ void conv_wmma_kernel(const float* __restrict__ X,    // [B,S,D]
                                 const float* __restrict__ Wt,   // [K,D/2,F,2]
                                 const float* __restrict__ bias, // [F]
                                 float* __restrict__ Y) {        // [B,S,F]
    int b    = blockIdx.z;
    int s0   = blockIdx.x * 16;
    int f0   = blockIdx.y * 64;
    int lane = threadIdx.x;
    int lm   = lane & 15;
    int kh   = lane >> 4;

    v8f acc[4] = {};

    const float* xb = X + (size_t)b * S_ * D_;

    #pragma unroll
    for (int kk = 0; kk < K_; ++kk) {
        int  row   = s0 + lm + kk - 1;
        bool valid = (row >= 0) && (row < S_);
        const float* xr = xb + (size_t)(valid ? row : 0) * D_;
        // lane's pair base within repacked weights for this kk:
        // element [kk][d2][f][0..1], with d2 = d0/2 + kh, f = f0 + lm
        const float* wk = Wt + (((size_t)kk * (D_ / 2) + kh) * F_ + f0 + lm) * 2;

        for (int d0 = 0; d0 < D_; d0 += 4) {
            int dd = d0 + 2 * kh;
            v2f a = {0.f, 0.f};
            if (valid) a = *(const v2f*)(xr + dd);

            const float* wrow = wk + (size_t)(d0 / 2) * (F_ * 2);
            #pragma unroll
            for (int j = 0; j < 4; ++j) {
                v2f bm = *(const v2f*)(wrow + 32 * j);   // {B[dd,f], B[dd+1,f]}
                acc[j] = __builtin_amdgcn_wmma_f32_16x16x4_f32(
                    false, a, false, bm, (short)0, acc[j], false, false);
            }
        }
    }

    float* yb = Y + ((size_t)b * S_ + s0) * F_ + f0;
    #pragma unroll
    for (int j = 0; j < 4; ++j) {
        float bv = bias[f0 + 16 * j + lm];
        #pragma unroll
        for (int i = 0; i < 8; ++i) {
            int r = i + 8 * kh;
            yb[(size_t)r * F_ + 16 * j + lm] = acc[j][i] + bv;
        }
    }
}

// ---------------------------------------------------------------------------
// In-place LayerNorm + ReLU over feature dim (256), one wave per (b,s) row.
// Optionally fuses reg_len = relu(h . wl + bl).
// ---------------------------------------------------------------------------
__global__ void ln_relu_kernel(float* __restrict__ H,            // [B*S, F]
                               const float* __restrict__ g,
                               const float* __restrict__ be,
                               const float* __restrict__ wl,     // [F] (or null)
                               const float* __restrict__ blp,    // [1] (or null)
                               float* __restrict__ reg_len,      // [B*S] (or null)
                               int computeLen) {
    int wave = threadIdx.x >> 5;
    int lane = threadIdx.x & 31;
    int row  = blockIdx.x * 8 + wave;                // 0 .. B*S-1
    float* hr = H + (size_t)row * F_;

    float4 v0 = *(const float4*)(hr + lane * 8);
    float4 v1 = *(const float4*)(hr + lane * 8 + 4);

    float s  = v0.x + v0.y + v0.z + v0.w + v1.x + v1.y + v1.z + v1.w;
    float ss = v0.x * v0.x + v0.y * v0.y + v0.z * v0.z + v0.w * v0.w +
               v1.x * v1.x + v1.y * v1.y + v1.z * v1.z + v1.w * v1.w;
    #pragma unroll
    for (int o = 16; o > 0; o >>= 1) {
        s  += __shfl_xor(s, o);
        ss += __shfl_xor(ss, o);
    }
    float mean = s * (1.0f / F_);
    float var  = ss * (1.0f / F_) - mean * mean;
    float inv  = rsqrtf(var + 1e-5f);

    float out[8] = {v0.x, v0.y, v0.z, v0.w, v1.x, v1.y, v1.z, v1.w};
    float dot = 0.f;
    #pragma unroll
    for (int j = 0; j < 8; ++j) {
        int fi = lane * 8 + j;
        float h = (out[j] - mean) * inv * g[fi] + be[fi];
        h = fmaxf(h, 0.0f);
        out[j] = h;
        if (computeLen) dot += h * wl[fi];
    }
    float4 w0  = {out[0], out[1], out[2], out[3]};
    float4 w1v = {out[4], out[5], out[6], out[7]};
    *(float4*)(hr + lane * 8)     = w0;
    *(float4*)(hr + lane * 8 + 4) = w1v;

    if (computeLen) {
        #pragma unroll
        for (int o = 16; o > 0; o >>= 1) dot += __shfl_xor(dot, o);
        if (lane == 0) reg_len[row] = fmaxf(dot + blp[0], 0.0f);
    }
}

// ---------------------------------------------------------------------------
// Per-batch inclusive cumsum of durations: one wave per batch.
// ---------------------------------------------------------------------------
__global__ void cumsum_kernel(const int* __restrict__ tgt, int* __restrict__ ends) {
    int b    = blockIdx.x;
    int lane = threadIdx.x;
    const int* tb = tgt + b * S_;
    int* eb = ends + b * S_;

    int v[16];
    int run = 0;
    #pragma unroll
    for (int i = 0; i < 16; ++i) { run += tb[lane * 16 + i]; v[i] = run; }

    int inc = run;
    #pragma unroll
    for (int o = 1; o < 32; o <<= 1) {
        int n = __shfl_up(inc, o);
        if (lane >= o) inc += n;
    }
    int ex = inc - run;
    #pragma unroll
    for (int i = 0; i < 16; ++i) eb[lane * 16 + i] = v[i] + ex;
}

// ---------------------------------------------------------------------------
// Length-regulator expansion: out[b,t,:] = x[b,s,:] where ends[s-1]<=t<ends[s];
// zeros past total length. 4 t-rows per block, float4 per thread.
// ---------------------------------------------------------------------------
__global__ void expand_kernel(const float* __restrict__ X,
                              const int* __restrict__ ends,
                              float* __restrict__ Out, int T) {
    int b  = blockIdx.y;
    int t  = blockIdx.x * 4 + (threadIdx.x >> 6);
    int d4 = (threadIdx.x & 63) * 4;
    if (t >= T) return;

    const int* eb = ends + b * S_;
    int total = eb[S_ - 1];

    float4 out = {0.f, 0.f, 0.f, 0.f};
    if (t < total) {
        int lo = 0, hi = S_ - 1;
        while (lo < hi) {
            int mid = (lo + hi) >> 1;
            if (eb[mid] > t) hi = mid; else lo = mid + 1;
        }
        out = *(const float4*)(X + ((size_t)b * S_ + lo) * D_ + d4);
    }
    *(float4*)(Out + ((size_t)b * T + t) * D_ + d4) = out;
}

// ---------------------------------------------------------------------------
extern "C" void kernel_launch(void* const* d_in, const int* in_sizes, int n_in,
                              void* d_out, int out_size, void* d_ws, size_t ws_size,
                              hipStream_t stream) {
    const float* x      = (const float*)d_in[0];
    const int*   target = (const int*)  d_in[1];
    const float* w1     = (const float*)d_in[2];
    const float* b1     = (const float*)d_in[3];
    const float* w2     = (const float*)d_in[4];
    const float* b2     = (const float*)d_in[5];
    const float* g1     = (const float*)d_in[6];
    const float* be1    = (const float*)d_in[7];
    const float* g2     = (const float*)d_in[8];
    const float* be2    = (const float*)d_in[9];
    const float* wl     = (const float*)d_in[10];
    const float* bl     = (const float*)d_in[11];

    // out = concat( [B,T,D] expanded, [B,S] reg_len )
    int T = (out_size - B_ * S_) / (B_ * D_);

    float* ws   = (float*)d_ws;
    float* W1t  = ws;                                  // K*D*F
    float* W2t  = W1t + (size_t)K_ * D_ * F_;
    float* buf1 = W2t + (size_t)K_ * D_ * F_;          // B*S*F (conv1 out / h1, in-place LN)
    float* buf2 = buf1 + (size_t)B_ * S_ * F_;         // B*S*F (conv2 out / h2)
    int*   ends = (int*)(buf2 + (size_t)B_ * S_ * F_); // B*S

    float* out_expand = (float*)d_out;
    float* reg_len    = out_expand + (size_t)B_ * T * D_;

    int wtot = K_ * D_ * F_;
    repack_w_kernel<<<(wtot + 255) / 256, 256, 0, stream>>>(w1, W1t);
    repack_w_kernel<<<(wtot + 255) / 256, 256, 0, stream>>>(w2, W2t);

    dim3 cgrid(S_ / 16, F_ / 64, B_);
    conv_wmma_kernel<<<cgrid, 32, 0, stream>>>(x, W1t, b1, buf1);
    ln_relu_kernel<<<B_ * S_ / 8, 256, 0, stream>>>(buf1, g1, be1, nullptr, nullptr, nullptr, 0);
    conv_wmma_kernel<<<cgrid, 32, 0, stream>>>(buf1, W2t, b2, buf2);
    ln_relu_kernel<<<B_ * S_ / 8, 256, 0, stream>>>(buf2, g2, be2, wl, bl, reg_len, 1);

    cumsum_kernel<<<B_, 32, 0, stream>>>(target, ends);
    if (T > 0) {
        dim3 egrid((T + 3) / 4, B_);
        expand_kernel<<<egrid, 256, 0, stream>>>(x, ends, out_expand, T);
    }
}